// GCN_67946382623222
// MI455X (gfx1250) — compile-verified
//
#include <hip/hip_runtime.h>

typedef __attribute__((ext_vector_type(2))) float v2f;
typedef __attribute__((ext_vector_type(8))) float v8f;

#define D_H 64   // hidden/output feature width (H1 = H2 = D_OUT = 64)

// ---------------------------------------------------------------- degrees ---
__global__ void k_init_deg(float* __restrict__ deg, int n) {
  int i = blockIdx.x * blockDim.x + threadIdx.x;
  if (i < n) deg[i] = 1.0f;                       // self-loop contributes 1
}

__global__ void k_count_deg(const int* __restrict__ dst, float* __restrict__ deg, int e) {
  int i = blockIdx.x * blockDim.x + threadIdx.x;
  if (i < e) atomicAdd(&deg[dst[i]], 1.0f);
}

__global__ void k_dinv(const float* __restrict__ deg, float* __restrict__ dinv, int n) {
  int i = blockIdx.x * blockDim.x + threadIdx.x;
  if (i < n) dinv[i] = rsqrtf(deg[i]);
}

// ------------------------------------------------------------- WMMA GEMM ----
// out[nrows x 64] = (RELU ? max(in,0) : in)[nrows x K] @ W[K x 64]
// One wave -> 16x64 output tile via V_WMMA_F32_16X16X4_F32 (fp32 matrix path).
// A 16x4 layout: lanes 0-15 hold (K=0,K=1), lanes 16-31 hold (K=2,K=3).
// C/D 16x16 layout: VGPR r -> M=r (lanes 0-15) / M=r+8 (lanes 16-31), N=lane&15.
//
// W is staged in LDS with K-pairs interleaved:
//   lds_w[(k/2)*128 + n*2 + (k&1)]  ==  W[k][n]
// so each lane's B operand (W[k][n], W[k+1][n]) is one contiguous 8-byte
// ds_load_b64 straight into an adjacent (even-aligned) VGPR pair — no movs.
template <int K, bool RELU>
__global__ __launch_bounds__(256) void k_gemm(const float* __restrict__ in,
                                              const float* __restrict__ W,
                                              float* __restrict__ out,
                                              int nrows) {
  __shared__ float lds_w[K * D_H];
  for (int i = threadIdx.x; i < K * D_H; i += 256) {
    int k = i / D_H;
    int n = i - k * D_H;
    lds_w[(k >> 1) * (2 * D_H) + n * 2 + (k & 1)] = W[i];
  }
  __syncthreads();

  const int wave = threadIdx.x >> 5;
  const int lane = threadIdx.x & 31;
  const int half = lane >> 4;        // 0: K pair {k0,k0+1}, 1: K pair {k0+2,k0+3}
  const int l16  = lane & 15;
  const int m0   = blockIdx.x * 128 + wave * 16;
  if (m0 >= nrows) return;           // whole-wave exit; EXEC stays all-1s for WMMA

  const float* __restrict__ arow = in + (size_t)(m0 + l16) * K + 2 * half;
  // per-lane LDS base: pair row (half) within each k0 group, column l16
  const float* __restrict__ wbase = &lds_w[half * (2 * D_H) + l16 * 2];

  v8f acc[4] = {v8f{}, v8f{}, v8f{}, v8f{}};

#pragma unroll
  for (int k0 = 0; k0 < K; k0 += 4) {
    float a0 = arow[k0];
    float a1 = arow[k0 + 1];
    if (RELU) { a0 = fmaxf(a0, 0.0f); a1 = fmaxf(a1, 0.0f); }
    v2f a; a.x = a0; a.y = a1;

    const float* __restrict__ wk = wbase + (k0 >> 1) * (2 * D_H);
#pragma unroll
    for (int j = 0; j < 4; ++j) {
      v2f b = *(const v2f*)(wk + j * 32);   // ds_load_b64, contiguous pair
      acc[j] = __builtin_amdgcn_wmma_f32_16x16x4_f32(
          /*neg_a=*/false, a, /*neg_b=*/false, b,
          /*c_mod=*/(short)0, acc[j], /*reuse_a=*/false, /*reuse_b=*/false);
    }
  }

  float* __restrict__ orow = out + (size_t)m0 * D_H;
#pragma unroll
  for (int j = 0; j < 4; ++j) {
#pragma unroll
    for (int r = 0; r < 8; ++r) {
      const int row = r + half * 8;
      orow[(size_t)row * D_H + j * 16 + l16] = acc[j][r];
    }
  }
}

// ------------------------------------------------- self-loop + bias (init) --
// agg = dinv^2 * t + b    (full overwrite -> also initializes the buffer)
__global__ void k_selfbias(const float* __restrict__ t, const float* __restrict__ dinv,
                           const float* __restrict__ b, float* __restrict__ agg, int n) {
  int idx = blockIdx.x * blockDim.x + threadIdx.x;
  if (idx >= n * D_H) return;
  int i = idx >> 6;
  int c = idx & 63;
  float dv = dinv[i];
  agg[idx] = dv * dv * t[idx] + b[c];
}

// ---------------------------------------------------------- edge scatter ----
// agg[dst] += dinv[src]*dinv[dst] * t[src]; 16 lanes x float4 per edge.
__global__ void k_scatter(const float* __restrict__ t, const int* __restrict__ src,
                          const int* __restrict__ dst, const float* __restrict__ dinv,
                          float* __restrict__ agg, int e) {
  int idx = blockIdx.x * blockDim.x + threadIdx.x;
  if (idx >= e * 16) return;
  int eid = idx >> 4;
  int c   = (idx & 15) * 4;
  int s = src[eid];
  int d = dst[eid];
  float w = dinv[s] * dinv[d];
  const float4 v = *(const float4*)(t + (size_t)s * D_H + c);
  float* ap = agg + (size_t)d * D_H + c;
  atomicAdd(ap + 0, w * v.x);
  atomicAdd(ap + 1, w * v.y);
  atomicAdd(ap + 2, w * v.z);
  atomicAdd(ap + 3, w * v.w);
}

// -------------------------------------------------------------- launcher ----
extern "C" void kernel_launch(void* const* d_in, const int* in_sizes, int n_in,
                              void* d_out, int out_size, void* d_ws, size_t ws_size,
                              hipStream_t stream) {
  const float* x  = (const float*)d_in[0];
  const float* W1 = (const float*)d_in[1];
  const float* b1 = (const float*)d_in[2];
  const float* W2 = (const float*)d_in[3];
  const float* b2 = (const float*)d_in[4];
  const float* W3 = (const float*)d_in[5];
  const float* b3 = (const float*)d_in[6];
  const int*   ei = (const int*)d_in[7];

  const int N = in_sizes[0] / 128;   // x is [N, 128]
  const int E = in_sizes[7] / 2;     // edge_index is [2, E]
  const int* src = ei;
  const int* dst = ei + E;

  float* ws   = (float*)d_ws;
  float* deg  = ws;
  float* dinv = ws + N;
  float* t    = ws + 2 * (size_t)N;            // [N, 64] transformed features
  float* agg  = t + (size_t)N * D_H;           // [N, 64] aggregation buffer
  float* out  = (float*)d_out;

  const int TB = 256;
  const int nb_n  = (N + TB - 1) / TB;
  const int nb_e  = (E + TB - 1) / TB;
  const int nb_nc = (N * D_H + TB - 1) / TB;
  const int nb_sc = (E * 16 + TB - 1) / TB;
  const int nb_g  = (N + 127) / 128;

  // normalization: deg = 1 + in-degree(dst); dinv = rsqrt(deg)
  k_init_deg <<<nb_n, TB, 0, stream>>>(deg, N);
  k_count_deg<<<nb_e, TB, 0, stream>>>(dst, deg, E);
  k_dinv     <<<nb_n, TB, 0, stream>>>(deg, dinv, N);

  // layer 1: t = x @ W1 ; agg = dinv^2*t + b1 + scatter
  k_gemm<128, false><<<nb_g, TB, 0, stream>>>(x, W1, t, N);
  k_selfbias<<<nb_nc, TB, 0, stream>>>(t, dinv, b1, agg, N);
  k_scatter <<<nb_sc, TB, 0, stream>>>(t, src, dst, dinv, agg, E);

  // layer 2: relu fused into GEMM A-load
  k_gemm<64, true><<<nb_g, TB, 0, stream>>>(agg, W2, t, N);
  k_selfbias<<<nb_nc, TB, 0, stream>>>(t, dinv, b2, agg, N);
  k_scatter <<<nb_sc, TB, 0, stream>>>(t, src, dst, dinv, agg, E);

  // layer 3: aggregate straight into d_out (no final relu)
  k_gemm<64, true><<<nb_g, TB, 0, stream>>>(agg, W3, t, N);
  k_selfbias<<<nb_nc, TB, 0, stream>>>(t, dinv, b3, out, N);
  k_scatter <<<nb_sc, TB, 0, stream>>>(t, src, dst, dinv, out, E);
}